// H2GCN_63677185130712
// MI455X (gfx1250) — compile-verified
//
#include <hip/hip_runtime.h>

#define NFEAT 512
#define HID   64
#define CLS   16
#define CAT   448   // 7*HID concat width

typedef float v2f __attribute__((ext_vector_type(2)));
typedef float v8f __attribute__((ext_vector_type(8)));

// ---------------- zero the concat workspace ----------------
__global__ void zero_ws_kernel(float4* __restrict__ p, long n4) {
    long i = (long)blockIdx.x * blockDim.x + threadIdx.x;
    long stride = (long)gridDim.x * blockDim.x;
    float4 z; z.x = z.y = z.z = z.w = 0.f;
    for (; i < n4; i += stride) p[i] = z;
}

// ---------------- R[:,0:64] = relu(x @ w_embed), WMMA f32 16x16x4 ----------------
// block = 128 threads (4 waves). Block owns 16 rows; wave w owns columns [16w,16w+16).
__global__ void embed_gemm_wmma(const float* __restrict__ x,
                                const float* __restrict__ w,
                                float* __restrict__ R) {
    const int lane = threadIdx.x & 31;
    const int wid  = threadIdx.x >> 5;
    const int hi   = lane >> 4;        // half-wave selector (K pair)
    const int l16  = lane & 15;
    const int m0   = blockIdx.x * 16;
    const int n0   = wid * 16;
    const float* xr = x + (size_t)(m0 + l16) * NFEAT;
    v8f c = {};
    for (int k = 0; k < NFEAT; k += 4) {
        const int ka = k + 2 * hi;
        v2f a, b;
        a.x = xr[ka];
        a.y = xr[ka + 1];
        b.x = w[(size_t)ka       * HID + n0 + l16];
        b.y = w[(size_t)(ka + 1) * HID + n0 + l16];
        c = __builtin_amdgcn_wmma_f32_16x16x4_f32(false, a, false, b,
                                                  (short)0, c, false, false);
    }
#pragma unroll
    for (int v = 0; v < 8; ++v) {
        const int m = m0 + v + 8 * hi;
        R[(size_t)m * CAT + n0 + l16] = fmaxf(c[v], 0.f);
    }
}

// ---------------- edge-parallel SpMM scatter (L2-resident atomics) ----------------
// D lanes per edge; a wave's 32 lanes cover 128 contiguous bytes of one source row.
template <int D, int LOGD, bool RELU_SRC>
__global__ void spmm_scatter(const int*   __restrict__ rows,
                             const int*   __restrict__ cols,
                             const float* __restrict__ val, int E,
                             const float* __restrict__ src,   // + column offset, stride CAT
                             float*       __restrict__ dst) { // + column offset, stride CAT
    const long t = (long)blockIdx.x * blockDim.x + threadIdx.x;
    const int  e = (int)(t >> LOGD);
    const int  f = (int)(t & (D - 1));
    if (e >= E) return;
    const int   r = rows[e];
    const int   c = cols[e];
    const float v = val[e];
    float h = src[(size_t)c * CAT + f];
    if (RELU_SRC) h = fmaxf(h, 0.f);
    atomicAdd(dst + (size_t)r * CAT + f, v * h);   // global_atomic_add_f32, no return
}

// ---------------- logits = concat @ w_classify (relu fused on cols >= 64) ----------------
// block = 128 threads (4 waves), each wave computes one 16x16 tile (16 rows x 16 classes).
__global__ void classify_gemm_wmma(const float* __restrict__ R,
                                   const float* __restrict__ wc,
                                   float* __restrict__ out, int ntiles) {
    const int lane = threadIdx.x & 31;
    const int wid  = threadIdx.x >> 5;
    const int tile = blockIdx.x * 4 + wid;
    if (tile >= ntiles) return;       // wave-uniform: EXEC stays all-ones for WMMA
    const int hi  = lane >> 4;
    const int l16 = lane & 15;
    const int m0  = tile * 16;
    const float* ar = R + (size_t)(m0 + l16) * CAT;
    v8f c = {};
    for (int k = 0; k < CAT; k += 4) {
        const int ka = k + 2 * hi;
        float a0 = ar[ka];
        float a1 = ar[ka + 1];
        if (k >= HID) { a0 = fmaxf(a0, 0.f); a1 = fmaxf(a1, 0.f); }  // relu on hop outputs
        v2f a, b;
        a.x = a0; a.y = a1;
        b.x = wc[(size_t)ka       * CLS + l16];
        b.y = wc[(size_t)(ka + 1) * CLS + l16];
        c = __builtin_amdgcn_wmma_f32_16x16x4_f32(false, a, false, b,
                                                  (short)0, c, false, false);
    }
#pragma unroll
    for (int v = 0; v < 8; ++v)
        out[(size_t)(m0 + v + 8 * hi) * CLS + l16] = c[v];
}

// ---------------- in-place row-wise log_softmax over 16 classes ----------------
__global__ void log_softmax_kernel(float* __restrict__ out, int N) {
    const int r = blockIdx.x * blockDim.x + threadIdx.x;
    if (r >= N) return;
    float* p = out + (size_t)r * CLS;
    float buf[CLS];
    float mx = -INFINITY;
#pragma unroll
    for (int i = 0; i < CLS; ++i) { buf[i] = p[i]; mx = fmaxf(mx, buf[i]); }
    float s = 0.f;
#pragma unroll
    for (int i = 0; i < CLS; ++i) s += __expf(buf[i] - mx);
    const float l = __logf(s);
#pragma unroll
    for (int i = 0; i < CLS; ++i) p[i] = buf[i] - mx - l;
}

extern "C" void kernel_launch(void* const* d_in, const int* in_sizes, int n_in,
                              void* d_out, int out_size, void* d_ws, size_t ws_size,
                              hipStream_t stream) {
    const float* x       = (const float*)d_in[0];
    const int*   a1_idx  = (const int*)  d_in[1];   // [2,E1]: rows then cols
    const float* a1_val  = (const float*)d_in[2];
    const int*   a2_idx  = (const int*)  d_in[3];
    const float* a2_val  = (const float*)d_in[4];
    const float* w_embed = (const float*)d_in[5];
    const float* w_cls   = (const float*)d_in[6];
    const int E1 = in_sizes[2];
    const int E2 = in_sizes[4];
    const int N  = in_sizes[0] / NFEAT;
    float* R = (float*)d_ws;                        // [N, 448] concat workspace
    float* logits = (float*)d_out;                  // [N, 16]

    // 1) zero concat workspace (atomic accumulation targets)
    const long n4 = (long)N * CAT / 4;
    zero_ws_kernel<<<4096, 256, 0, stream>>>((float4*)R, n4);

    // 2) r0 = relu(x @ w_embed) -> R[:,0:64]
    embed_gemm_wmma<<<N / 16, 128, 0, stream>>>(x, w_embed, R);

    // 3) hop 1: src cols [0:64), D=64 lanes/edge (operand 25.6MB -> L2 resident)
    {
        long t1 = (long)E1 * 64;
        spmm_scatter<64, 6, false><<<(int)((t1 + 255) / 256), 256, 0, stream>>>(
            a1_idx, a1_idx + E1, a1_val, E1, R, R + HID);          // -> cols [64:128)
        long t2 = (long)E2 * 64;
        spmm_scatter<64, 6, false><<<(int)((t2 + 255) / 256), 256, 0, stream>>>(
            a2_idx, a2_idx + E2, a2_val, E2, R, R + 2 * HID);      // -> cols [128:192)
    }

    // 4) hop 2: src cols [64:192) with fused relu, D=128 lanes/edge
    {
        long t1 = (long)E1 * 128;
        spmm_scatter<128, 7, true><<<(int)((t1 + 255) / 256), 256, 0, stream>>>(
            a1_idx, a1_idx + E1, a1_val, E1, R + HID, R + 3 * HID); // -> cols [192:320)
        long t2 = (long)E2 * 128;
        spmm_scatter<128, 7, true><<<(int)((t2 + 255) / 256), 256, 0, stream>>>(
            a2_idx, a2_idx + E2, a2_val, E2, R + HID, R + 5 * HID); // -> cols [320:448)
    }

    // 5) logits = relu-fused concat @ w_classify
    const int ntiles = N / 16;
    classify_gemm_wmma<<<(ntiles + 3) / 4, 128, 0, stream>>>(R, w_cls, logits, ntiles);

    // 6) in-place log_softmax
    log_softmax_kernel<<<(N + 255) / 256, 256, 0, stream>>>(logits, N);
}